// LinearCRFWordSegmentation_33904471834764
// MI455X (gfx1250) — compile-verified
//
#include <hip/hip_runtime.h>
#include <cstdint>

// Problem constants (match reference file)
#define BB 8192
#define LL 1024
#define TT 4

#define LOG2E 1.4426950408889634f
#define LN2   0.6931471805599453f

#define NW0   512          // chunk-0 waves (16 batches each)
#define NWMAT 2048         // waves per matrix chunk (4 batches each)

typedef float v2f __attribute__((ext_vector_type(2)));
typedef float v8f __attribute__((ext_vector_type(8)));

// ---------------------------------------------------------------------------
// ws layout (32-bit words):
//  [0] score acc (f)   [1] logZ acc (f)   [2] fallback flag (i)   [3] pad
//  [4        .. 4+B  )  int   cnt[b]
//  [4+B      .. 4+2B )  int   end_pos[b]
//  [4+2B     .. 4+6B )  float alphaA[b][4]        (chunk0 result / slow-path cap)
//  [4+6B     .. 4+7B )  float scaleA[b]
//  [4+7B     .. 4+55B)  float P[c][b][4][4]       (3 chunk composites)
//  [4+55B    .. 4+67B)  float scaleC[c][b][4]     (per-row scales)
// ---------------------------------------------------------------------------
#define OFF_CNT    4
#define OFF_ENDP   (4 + BB)
#define OFF_ALPHA  (4 + 2 * BB)
#define OFF_SCALEA (4 + 6 * BB)
#define OFF_P      (4 + 7 * BB)
#define OFF_SCALEC (4 + 55 * BB)

__global__ void crf_init_ws(float* ws) {
    const int i = blockIdx.x * blockDim.x + threadIdx.x;
    if (i < 4 + BB) ws[i] = 0.0f;   // score, logZ, flag, pad, cnt[]
}

// Identity cross between lane halves (lane i <-> lane i^16), pure VALU.
static __device__ __forceinline__ float permx16(float x) {
    union { float f; unsigned u; } c;
    c.f = x;
    c.u = __builtin_amdgcn_permlanex16(c.u, c.u, 0x76543210u, 0xfedcba98u, false, false);
    return c.f;
}

static __device__ __forceinline__ v8f wmma4(v2f a, v2f b) {
    v8f c = {};
    return __builtin_amdgcn_wmma_f32_16x16x4_f32(
        false, a, false, b, (short)0, c, false, false);
}

// ---------------------------------------------------------------------------
// Gold path score, pass 1: 4 segments per row for 4x memory-level parallelism
// on the dependent tag->emit gather chain; also warms L2 with s_tag.
// ---------------------------------------------------------------------------
__global__ void crf_gold_seg(const float* __restrict__ s_tag,
                             const int*   __restrict__ tags,
                             const unsigned char* __restrict__ mask, // jnp bool = 1B
                             const float* __restrict__ trans,
                             const float* __restrict__ start,
                             float* __restrict__ score_acc,
                             int*   __restrict__ cnt) {
    const int seg = blockIdx.x >> 5;                        // 0..3
    const int b   = ((blockIdx.x & 31) << 8) + threadIdx.x; // 0..8191
    const int l0  = seg << 8;

    const int*           tg = tags + (size_t)b * LL;
    const unsigned char* mk = mask + (size_t)b * LL;
    const float*         st = s_tag + (size_t)b * LL * TT;

    float acc = 0.0f;
    int   ccnt = 0;
    int   prev = (l0 > 0) ? tg[l0 - 1] : 0;
    #pragma unroll 4
    for (int l = l0; l < l0 + 256; ++l) {
        const int t = tg[l];
        const float emit = st[(size_t)l * TT + t];
        const float tr = (l > 0) ? trans[prev * TT + t] : 0.0f;
        if (mk[l] != 0) { acc += emit + tr; ccnt++; }
        prev = t;
    }
    if (seg == 0) acc += start[tg[0]];
    atomicAdd(&cnt[b], ccnt);      // 8192 distinct addresses: no hot spot

    __shared__ float red[256];
    red[threadIdx.x] = acc;
    __syncthreads();
    for (int s = 128; s > 0; s >>= 1) {
        if (threadIdx.x < s) red[threadIdx.x] += red[threadIdx.x + s];
        __syncthreads();
    }
    if (threadIdx.x == 0) atomicAdd(score_acc, red[0]);
}

// Gold pass 2: end_pos = cnt-1, add end_score; set fallback flag if any
// end_pos != L-1 (forces the monolithic capturing scan).
__global__ void crf_gold_end(const int* __restrict__ tags,
                             const float* __restrict__ endsc,
                             const int* __restrict__ cnt,
                             int* __restrict__ end_pos,
                             int* __restrict__ flag,
                             float* __restrict__ score_acc) {
    const int b = blockIdx.x * blockDim.x + threadIdx.x;
    int ep = cnt[b] - 1;
    if (ep < 0) ep = 0;
    end_pos[b] = ep;
    if (__any(ep != LL - 1) && (threadIdx.x & 31) == 0) atomicOr(flag, 1);
    float acc = endsc[tags[(size_t)b * LL + ep]];

    __shared__ float red[256];
    red[threadIdx.x] = acc;
    __syncthreads();
    for (int s = 128; s > 0; s >>= 1) {
        if (threadIdx.x < s) red[threadIdx.x] += red[threadIdx.x + s];
        __syncthreads();
    }
    if (threadIdx.x == 0) atomicAdd(score_acc, red[0]);
}

// ---------------------------------------------------------------------------
// Scan bodies. Recurrence per step: bv = D(0,1) * exp(emit); D = wmma(A,bv).
// scale2 carried in log2 units. Renorm once per 4-step group (fp32-safe).
// ---------------------------------------------------------------------------
static __device__ __forceinline__ void crf_step4(const float* __restrict__ p,
                                                 v2f a, v2f& bv, float& scale2) {
    const float2 e0 = *reinterpret_cast<const float2*>(p);
    const float2 e1 = *reinterpret_cast<const float2*>(p + TT);
    const float2 e2 = *reinterpret_cast<const float2*>(p + 2 * TT);
    const float2 e3 = *reinterpret_cast<const float2*>(p + 3 * TT);
    __builtin_prefetch(p + 16 * TT, 0, 3);

    const float x0 = __expf(e0.x), y0 = __expf(e0.y);
    const float x1 = __expf(e1.x), y1 = __expf(e1.y);
    const float x2 = __expf(e2.x), y2 = __expf(e2.y);
    const float x3 = __expf(e3.x), y3 = __expf(e3.y);

    v8f d;
    d = wmma4(a, bv); bv.x = d[0] * x0; bv.y = d[1] * y0;
    d = wmma4(a, bv); bv.x = d[0] * x1; bv.y = d[1] * y1;
    d = wmma4(a, bv); bv.x = d[0] * x2; bv.y = d[1] * y2;
    d = wmma4(a, bv); bv.x = d[0] * x3; bv.y = d[1] * y3;

    float mm = fmaxf(bv.x, bv.y);
    mm = fmaxf(mm, permx16(mm));
    const float inv = __builtin_amdgcn_rcpf(mm);
    bv.x *= inv;
    bv.y *= inv;
    scale2 += __builtin_amdgcn_logf(mm);   // raw v_log_f32 (log2)
}

static __device__ __forceinline__ void crf_scan_fast(const float* __restrict__ row,
                                                     v2f a, int k0, int lbeg, int ngroups,
                                                     v2f& bv, float& scale2) {
    for (int g = 0; g < ngroups; ++g)
        crf_step4(row + (size_t)(lbeg + g * 4) * TT + k0, a, bv, scale2);
}

static __device__ __forceinline__ void crf_tail3(const float* __restrict__ row,
                                                 v2f a, int k0, int lbeg,
                                                 v2f& bv, float& scale2) {
    const float* p = row + (size_t)lbeg * TT + k0;
    const float2 e0 = *reinterpret_cast<const float2*>(p);
    const float2 e1 = *reinterpret_cast<const float2*>(p + TT);
    const float2 e2 = *reinterpret_cast<const float2*>(p + 2 * TT);
    const float x0 = __expf(e0.x), y0 = __expf(e0.y);
    const float x1 = __expf(e1.x), y1 = __expf(e1.y);
    const float x2 = __expf(e2.x), y2 = __expf(e2.y);
    v8f d;
    d = wmma4(a, bv); bv.x = d[0] * x0; bv.y = d[1] * y0;
    d = wmma4(a, bv); bv.x = d[0] * x1; bv.y = d[1] * y1;
    d = wmma4(a, bv); bv.x = d[0] * x2; bv.y = d[1] * y2;
}

// Slow path: full-length scan with end_pos capture (general mask).
static __device__ __forceinline__ void crf_scan_cap(const float* __restrict__ row,
                                                    v2f a, int k0, int endp,
                                                    v2f& bv, float& scale2,
                                                    float& cap0, float& cap1, float& capsc) {
#define CRF_CAP(L)                                       \
    {                                                    \
        const bool c_ = ((L) == endp);                   \
        cap0  = c_ ? bv.x   : cap0;                      \
        cap1  = c_ ? bv.y   : cap1;                      \
        capsc = c_ ? scale2 : capsc;                     \
    }
    for (int g = 0; g < (LL - 4) / 4; ++g) {
        const int l0 = 1 + g * 4;
        const float* p = row + (size_t)l0 * TT + k0;
        const float2 e0 = *reinterpret_cast<const float2*>(p);
        const float2 e1 = *reinterpret_cast<const float2*>(p + TT);
        const float2 e2 = *reinterpret_cast<const float2*>(p + 2 * TT);
        const float2 e3 = *reinterpret_cast<const float2*>(p + 3 * TT);
        const float x0 = __expf(e0.x), y0 = __expf(e0.y);
        const float x1 = __expf(e1.x), y1 = __expf(e1.y);
        const float x2 = __expf(e2.x), y2 = __expf(e2.y);
        const float x3 = __expf(e3.x), y3 = __expf(e3.y);
        v8f d;
        d = wmma4(a, bv); bv.x = d[0] * x0; bv.y = d[1] * y0; CRF_CAP(l0);
        d = wmma4(a, bv); bv.x = d[0] * x1; bv.y = d[1] * y1; CRF_CAP(l0 + 1);
        d = wmma4(a, bv); bv.x = d[0] * x2; bv.y = d[1] * y2; CRF_CAP(l0 + 2);
        d = wmma4(a, bv); bv.x = d[0] * x3; bv.y = d[1] * y3; CRF_CAP(l0 + 3);
        float mm = fmaxf(bv.x, bv.y);
        mm = fmaxf(mm, permx16(mm));
        const float inv = __builtin_amdgcn_rcpf(mm);
        bv.x *= inv; bv.y *= inv;
        scale2 += __builtin_amdgcn_logf(mm);
    }
    {   // tail l = LL-3..LL-1
        const int l0 = LL - 3;
        const float* p = row + (size_t)l0 * TT + k0;
        const float2 e0 = *reinterpret_cast<const float2*>(p);
        const float2 e1 = *reinterpret_cast<const float2*>(p + TT);
        const float2 e2 = *reinterpret_cast<const float2*>(p + 2 * TT);
        const float x0 = __expf(e0.x), y0 = __expf(e0.y);
        const float x1 = __expf(e1.x), y1 = __expf(e1.y);
        const float x2 = __expf(e2.x), y2 = __expf(e2.y);
        v8f d;
        d = wmma4(a, bv); bv.x = d[0] * x0; bv.y = d[1] * y0; CRF_CAP(l0);
        d = wmma4(a, bv); bv.x = d[0] * x1; bv.y = d[1] * y1; CRF_CAP(l0 + 1);
        d = wmma4(a, bv); bv.x = d[0] * x2; bv.y = d[1] * y2; CRF_CAP(l0 + 2);
    }
#undef CRF_CAP
}

// ---------------------------------------------------------------------------
// Forward scan, L split into 4 chunks of 256 (associative composition):
//  waves [0,512):      chunk0 vector scan, 16 batches/wave   (l = 0..255)
//  waves [512,6656):   chunks 1..3 composite 4x4 scans, 4 batches/wave
// Row-wise rescaling commutes through P·E·diag(x), so composite rows use the
// *identical* recurrence/renorm code, initialized to basis vectors.
// ---------------------------------------------------------------------------
__global__ void crf_forward_kernel(const float* __restrict__ s_tag,
                                   const float* __restrict__ trans,
                                   const float* __restrict__ start,
                                   const int*   __restrict__ end_pos,
                                   const int*   __restrict__ flagp,
                                   float* __restrict__ alphaA,
                                   float* __restrict__ scaleA,
                                   float* __restrict__ Pc,
                                   float* __restrict__ scaleC) {
    const int tid  = blockIdx.x * blockDim.x + threadIdx.x;
    const int wave = tid >> 5;
    const int lane = tid & 31;
    const int n    = lane & 15;
    const int half = lane >> 4;
    const int k0   = half * 2;
    const int k1   = k0 + 1;

    // A rows {0,1,8,9} = E columns for states {0,1,2,3}; D(0,1) == next B.
    const int sm = (n == 0) ? 0 : (n == 1) ? 1 : (n == 8) ? 2 : (n == 9) ? 3 : -1;
    v2f a;
    a.x = (sm >= 0) ? __expf(trans[k0 * TT + sm]) : 0.0f;
    a.y = (sm >= 0) ? __expf(trans[k1 * TT + sm]) : 0.0f;

    const int flag = flagp[0];

    if (flag == 0) {
        if (wave < NW0) {
            // ---- chunk 0: vector scan, batches wave*16+n, l = 0..255
            const int b = wave * 16 + n;
            const float* row = s_tag + (size_t)b * LL * TT;
            const float2 ei = *reinterpret_cast<const float2*>(row + k0);
            float s0 = ei.x + start[k0];
            float s1 = ei.y + start[k1];
            float mx = fmaxf(s0, s1);
            mx = fmaxf(mx, permx16(mx));
            v2f bv;
            bv.x = __expf(s0 - mx);
            bv.y = __expf(s1 - mx);
            float scale2 = mx * LOG2E;

            crf_scan_fast(row, a, k0, 1, 63, bv, scale2);     // l = 1..252
            crf_tail3(row, a, k0, 253, bv, scale2);           // l = 253..255

            *reinterpret_cast<float2*>(alphaA + (size_t)b * 4 + k0) =
                make_float2(bv.x, bv.y);
            if (half == 0) scaleA[b] = scale2;
        } else {
            // ---- chunks 1..3: composite rows, 4 batches/wave
            const int wv  = wave - NW0;
            const int c   = wv >> 11;            // 0..2 -> chunk c+1
            const int wix = wv & (NWMAT - 1);
            const int bat = wix * 4 + (n >> 2);
            const int r   = n & 3;               // composite row index
            const float* row = s_tag + (size_t)bat * LL * TT;

            v2f bv;                              // P row r = basis e_r
            bv.x = (r == k0) ? 1.0f : 0.0f;
            bv.y = (r == k1) ? 1.0f : 0.0f;
            float scale2 = 0.0f;

            crf_scan_fast(row, a, k0, (c + 1) * 256, 64, bv, scale2);

            const size_t pb = ((size_t)c * BB + bat);
            *reinterpret_cast<float2*>(Pc + pb * 16 + r * 4 + k0) =
                make_float2(bv.x, bv.y);
            if (half == 0) scaleC[pb * 4 + r] = scale2;
        }
    } else {
        // ---- general end_pos fallback: monolithic capturing scan
        if (wave >= NW0) return;
        const int b = wave * 16 + n;
        const float* row = s_tag + (size_t)b * LL * TT;
        const float2 ei = *reinterpret_cast<const float2*>(row + k0);
        float s0 = ei.x + start[k0];
        float s1 = ei.y + start[k1];
        float mx = fmaxf(s0, s1);
        mx = fmaxf(mx, permx16(mx));
        v2f bv;
        bv.x = __expf(s0 - mx);
        bv.y = __expf(s1 - mx);
        float scale2 = mx * LOG2E;
        const int endp = end_pos[b];
        float cap0 = bv.x, cap1 = bv.y, capsc = scale2;
        crf_scan_cap(row, a, k0, endp, bv, scale2, cap0, cap1, capsc);
        *reinterpret_cast<float2*>(alphaA + (size_t)b * 4 + k0) =
            make_float2(cap0, cap1);
        if (half == 0) scaleA[b] = capsc;
    }
}

// Combine: alpha · P1 · P2 · P3 with per-row log2 scales, then logZ reduce.
__global__ void crf_combine(const float* __restrict__ endsc,
                            const int*   __restrict__ flagp,
                            const float* __restrict__ alphaA,
                            const float* __restrict__ scaleA,
                            const float* __restrict__ Pc,
                            const float* __restrict__ scaleC,
                            float* __restrict__ logz_acc) {
    const int b = blockIdx.x * blockDim.x + threadIdx.x;
    float4 va = *reinterpret_cast<const float4*>(alphaA + (size_t)b * 4);
    float v0 = va.x, v1 = va.y, v2 = va.z, v3 = va.w;
    float s = scaleA[b];

    if (flagp[0] == 0) {
        #pragma unroll
        for (int c = 0; c < 3; ++c) {
            const size_t pb = ((size_t)c * BB + b);
            const float4 sr = *reinterpret_cast<const float4*>(scaleC + pb * 4);
            const float smax = fmaxf(fmaxf(sr.x, sr.y), fmaxf(sr.z, sr.w));
            const float t0 = v0 * __builtin_amdgcn_exp2f(sr.x - smax);
            const float t1 = v1 * __builtin_amdgcn_exp2f(sr.y - smax);
            const float t2 = v2 * __builtin_amdgcn_exp2f(sr.z - smax);
            const float t3 = v3 * __builtin_amdgcn_exp2f(sr.w - smax);
            const float* P = Pc + pb * 16;
            const float4 r0 = *reinterpret_cast<const float4*>(P);
            const float4 r1 = *reinterpret_cast<const float4*>(P + 4);
            const float4 r2 = *reinterpret_cast<const float4*>(P + 8);
            const float4 r3 = *reinterpret_cast<const float4*>(P + 12);
            float n0 = t0 * r0.x + t1 * r1.x + t2 * r2.x + t3 * r3.x;
            float n1 = t0 * r0.y + t1 * r1.y + t2 * r2.y + t3 * r3.y;
            float n2 = t0 * r0.z + t1 * r1.z + t2 * r2.z + t3 * r3.z;
            float n3 = t0 * r0.w + t1 * r1.w + t2 * r2.w + t3 * r3.w;
            const float mv = fmaxf(fmaxf(n0, n1), fmaxf(n2, n3));
            const float inv = __builtin_amdgcn_rcpf(mv);
            v0 = n0 * inv; v1 = n1 * inv; v2 = n2 * inv; v3 = n3 * inv;
            s += smax + __builtin_amdgcn_logf(mv);
        }
    }

    const float f = v0 * __expf(endsc[0]) + v1 * __expf(endsc[1]) +
                    v2 * __expf(endsc[2]) + v3 * __expf(endsc[3]);
    const float lz = (s + __builtin_amdgcn_logf(f)) * LN2;

    __shared__ float red[256];
    red[threadIdx.x] = lz;
    __syncthreads();
    for (int st = 128; st > 0; st >>= 1) {
        if (threadIdx.x < st) red[threadIdx.x] += red[threadIdx.x + st];
        __syncthreads();
    }
    if (threadIdx.x == 0) atomicAdd(logz_acc, red[0]);
}

__global__ void crf_finalize(const float* __restrict__ ws, float* __restrict__ out) {
    out[0] = (ws[1] - ws[0]) / (float)BB;
}

extern "C" void kernel_launch(void* const* d_in, const int* in_sizes, int n_in,
                              void* d_out, int out_size, void* d_ws, size_t ws_size,
                              hipStream_t stream) {
    const float* s_tag = (const float*)d_in[0];
    const int*   tags  = (const int*)d_in[1];
    const unsigned char* mask = (const unsigned char*)d_in[2]; // jnp bool -> 1 byte
    const float* trans = (const float*)d_in[3];
    const float* start = (const float*)d_in[4];
    const float* endsc = (const float*)d_in[5];

    float* wsf = (float*)d_ws;
    int*   wsi = (int*)d_ws;
    float* score_acc = wsf + 0;
    float* logz_acc  = wsf + 1;
    int*   flag      = wsi + 2;
    int*   cnt       = wsi + OFF_CNT;
    int*   end_pos   = wsi + OFF_ENDP;
    float* alphaA    = wsf + OFF_ALPHA;
    float* scaleA    = wsf + OFF_SCALEA;
    float* Pc        = wsf + OFF_P;
    float* scaleC    = wsf + OFF_SCALEC;

    crf_init_ws<<<(4 + BB + 255) / 256, 256, 0, stream>>>(wsf);

    crf_gold_seg<<<128, 256, 0, stream>>>(s_tag, tags, mask, trans, start,
                                          score_acc, cnt);
    crf_gold_end<<<BB / 256, 256, 0, stream>>>(tags, endsc, cnt, end_pos, flag,
                                               score_acc);

    // 512 + 3*2048 = 6656 waves -> 212992 threads -> 832 blocks
    crf_forward_kernel<<<(NW0 + 3 * NWMAT) / 8, 256, 0, stream>>>(
        s_tag, trans, start, end_pos, flag, alphaA, scaleA, Pc, scaleC);

    crf_combine<<<BB / 256, 256, 0, stream>>>(endsc, flag, alphaA, scaleA,
                                              Pc, scaleC, logz_acc);

    crf_finalize<<<1, 1, 0, stream>>>(wsf, (float*)d_out);
}